// NormalsRenderer_45749991637057
// MI455X (gfx1250) — compile-verified
//
#include <hip/hip_runtime.h>

typedef __attribute__((ext_vector_type(2))) float v2f;
typedef __attribute__((ext_vector_type(8))) float v8f;

// ---------------------------------------------------------------------------
// Phase 1: feats[p] = clip(colormap(flip(normals[p] @ w2v)))  -> float4 table
// One wave (32 threads) per 16-point tile, using V_WMMA_F32_16X16X4_F32:
//   A (16x4 f32): rows = 16 points, K = {nx, ny, nz, 0}
//   B (4x16 f32): cols 0..2 = w2v columns, rest zero
//   D (16x16)   : D[m][n] = nc[m][n] for n < 3
// ---------------------------------------------------------------------------
__global__ __launch_bounds__(32) void feats_kernel(const float* __restrict__ normals,
                                                   const float* __restrict__ w2v,
                                                   float4* __restrict__ feats,
                                                   int P) {
  const int lane = threadIdx.x;        // 0..31 (wave32)
  const int m    = lane & 15;          // row (A) / col (B) index within tile
  const int hi   = lane >> 4;          // selects K-half {0,1} vs {2,3}
  const int p    = blockIdx.x * 16 + m;
  const int pc   = (p < P) ? p : (P - 1);   // clamp loads; keep EXEC full for WMMA

  __shared__ float nc[16][3];

#if __has_builtin(__builtin_amdgcn_wmma_f32_16x16x4_f32)
  // ---- A operand: VGPR0 = K0 (lanes 0-15) / K2 (lanes 16-31); VGPR1 = K1 / K3(=0)
  const int offA = 3 * pc + (hi ? 2 : 0);
  float a0 = normals[offA];
  float a1 = 0.0f;
  if (!hi) a1 = normals[3 * pc + 1];

  // ---- B operand: lane%16 = N, VGPR0 = row K0/K2, VGPR1 = row K1/K3(=0)
  float b0 = 0.0f, b1 = 0.0f;
  if (m < 3) {
    b0 = w2v[(hi ? 2 : 0) * 3 + m];    // w2v[k][n] row-major
    if (!hi) b1 = w2v[3 + m];
  }

  v2f a; a[0] = a0; a[1] = a1;
  v2f b; b[0] = b0; b[1] = b1;
  v8f c = {};
  c = __builtin_amdgcn_wmma_f32_16x16x4_f32(false, a, false, b, (short)0, c,
                                            false, false);

  // D layout: VGPR r, lanes 0-15 -> D[r][lane]; lanes 16-31 -> D[8+r][lane-16]
  if (m < 3) {
#pragma unroll
    for (int r = 0; r < 8; ++r) nc[hi * 8 + r][m] = c[r];
  }
  __syncthreads();
#else
  // Scalar fallback: lanes 0..15 each do the 3x3 product directly.
  if (lane < 16) {
    float nx = normals[3 * pc + 0];
    float ny = normals[3 * pc + 1];
    float nz = normals[3 * pc + 2];
#pragma unroll
    for (int n = 0; n < 3; ++n)
      nc[m][n] = nx * w2v[0 * 3 + n] + ny * w2v[1 * 3 + n] + nz * w2v[2 * 3 + n];
  }
  __syncthreads();
#endif

  if (lane < 16 && p < P) {
    float x = nc[m][0], y = nc[m][1], z = nc[m][2];
    if (z > 0.0f) { x = -x; y = -y; z = -z; }            // flip toward camera
    float fx = __saturatef(x * 0.5f + 0.5f);             // clip to [0,1]
    float fy = __saturatef(0.5f - y * 0.5f);             // 1 - (y*0.5+0.5)
    float fz = __saturatef(0.5f - z * 0.5f);
    feats[p] = make_float4(fx, fy, fz, 0.0f);
  }
}

// ---------------------------------------------------------------------------
// Phase 2 (HBM-bound): per pixel, first-slot select + table gather.
//   read  idx[pix*K + 0]  (stream, non-temporal: 64 MiB single-use)
//   gather feats[i0]      (1.6 MB table, L2-resident)
//   write out[pix*3 ..+2] (stream, non-temporal: 24 MiB)
// ---------------------------------------------------------------------------
__global__ __launch_bounds__(256) void render_kernel(const int* __restrict__ idx,
                                                     const float4* __restrict__ feats,
                                                     float* __restrict__ out,
                                                     int npix, int K) {
  int pix = blockIdx.x * blockDim.x + threadIdx.x;
  if (pix >= npix) return;

  int i0 = __builtin_nontemporal_load(idx + (size_t)pix * (size_t)K);

  float r, g, b;
  if (i0 < 0) {                       // background: no point in first slot
    r = 0.5f; g = 0.5f; b = 1.0f;
  } else {
    float4 f = feats[i0];             // L2-hot gather
    r = f.x; g = f.y; b = f.z;
  }

  size_t o = (size_t)pix * 3;
  __builtin_nontemporal_store(r, out + o + 0);
  __builtin_nontemporal_store(g, out + o + 1);
  __builtin_nontemporal_store(b, out + o + 2);
}

extern "C" void kernel_launch(void* const* d_in, const int* in_sizes, int n_in,
                              void* d_out, int out_size, void* d_ws, size_t ws_size,
                              hipStream_t stream) {
  const int*   idx     = (const int*)d_in[0];     // [N,H,W,K] int32
  const float* normals = (const float*)d_in[1];   // [P,3] f32
  const float* w2v     = (const float*)d_in[2];   // [3,3] f32

  float* out = (float*)d_out;                     // [N,H,W,3] f32
  const int P    = in_sizes[1] / 3;
  const int npix = out_size / 3;
  const int K    = in_sizes[0] / npix;

  float4* feats = (float4*)d_ws;                  // P * 16 bytes scratch

  const int tiles = (P + 15) / 16;
  feats_kernel<<<tiles, 32, 0, stream>>>(normals, w2v, feats, P);

  render_kernel<<<(npix + 255) / 256, 256, 0, stream>>>(idx, feats, out, npix, K);
}